// SpatioTemporalBlock_48567490183326
// MI455X (gfx1250) — compile-verified
//
#include <hip/hip_runtime.h>
#include <hip/hip_bf16.h>

typedef float v2f __attribute__((ext_vector_type(2)));
typedef float v8f __attribute__((ext_vector_type(8)));

#define NNODE 7936
#define CIN   5
#define WLEN  64
#define T1    57
#define T2    50
#define H1C   64
#define H2C   16
#define NEDGE 63488

// padded LDS strides (64 banks x 4B): avoid stride % 64 == 0 patterns
#define XS_STRIDE 321   // per-node stride for x tile   (321 % 64 == 1)
#define HS_STRIDE 66    // per-row stride for GLU chunk (66*8 % 64 == 16)
#define GS_STRIDE 17    // per-node stride for G tile   (17*l % 64 distinct)

__device__ __forceinline__ v8f wmma_f32(v2f a, v2f b, v8f c) {
  // V_WMMA_F32_16X16X4_F32 : D(16x16,f32) = A(16x4,f32) x B(4x16,f32) + C
  return __builtin_amdgcn_wmma_f32_16x16x4_f32(false, a, false, b, (short)0, c,
                                               false, false);
}

__device__ __forceinline__ float sigmoidf_(float x) {
  return 1.0f / (1.0f + __expf(-x));
}

// ---------------------------------------------------------------------------
// init: gout = 0, deg = 1 (self loop weight)
// ---------------------------------------------------------------------------
__global__ void k_init(float* __restrict__ gout, float* __restrict__ deg, int gsz) {
  int i = blockIdx.x * blockDim.x + threadIdx.x;
  if (i < gsz) gout[i] = 0.0f;
  if (i < NNODE) deg[i] = 1.0f;
}

// ---------------------------------------------------------------------------
// fused: conv1 (K=40 GEMM, WMMA f32) -> GLU (in-register P/Q pairing)
//        -> Wg transform (K=64 GEMM, WMMA f32) -> xt[t][n][16]
// one block = 16 nodes, 256 threads = 8 waves
// ---------------------------------------------------------------------------
__global__ __launch_bounds__(256) void k_conv1(const float* __restrict__ x,
                                               const float* __restrict__ Wc1,
                                               const float* __restrict__ b1,
                                               const float* __restrict__ Wg,
                                               float* __restrict__ xt) {
  __shared__ float Xs[16 * XS_STRIDE];       // 20.1 KB : x tile [n][ci*64+w]
  __shared__ float Hs[8 * 16 * HS_STRIDE];   // 33 KB   : GLU chunk [tl][n][ch]

  const int tid    = threadIdx.x;
  const int lane   = tid & 31;
  const int wave   = tid >> 5;
  const int lane16 = lane & 15;
  const int half   = lane >> 4;
  const int n0     = blockIdx.x * 16;

  for (int idx = tid; idx < 16 * CIN * WLEN; idx += 256) {
    int n = idx / (CIN * WLEN), r = idx % (CIN * WLEN);
    Xs[n * XS_STRIDE + r] = x[(n0 + n) * (CIN * WLEN) + r];
  }

  // conv1 weight fragments: B[k][co] = Wc1[co*40 + k], tiles P (cb=pb) / Q (cb=pb+4)
  const int pb  = wave & 3;
  const int coP = pb * 16 + lane16;
  const int coQ = coP + 64;
  v2f bp1[10], bq1[10];
#pragma unroll
  for (int s = 0; s < 10; ++s) {
    int k     = 4 * s + 2 * half;
    bp1[s][0] = Wc1[coP * 40 + k];
    bp1[s][1] = Wc1[coP * 40 + k + 1];
    bq1[s][0] = Wc1[coQ * 40 + k];
    bq1[s][1] = Wc1[coQ * 40 + k + 1];
  }
  const float biasP = b1[coP];
  const float biasQ = b1[coQ];

  // Wg fragments: B[k][f] = Wg[k*16 + f]
  v2f bwg[16];
#pragma unroll
  for (int s = 0; s < 16; ++s) {
    int k     = 4 * s + 2 * half;
    bwg[s][0] = Wg[k * 16 + lane16];
    bwg[s][1] = Wg[(k + 1) * 16 + lane16];
  }
  __syncthreads();

  const int dt = wave >> 2;
  for (int tbase = 0; tbase < T1; tbase += 8) {
    // ---- phase 1: conv1 GEMM + GLU for 8 time steps (4 per wave) ----
    for (int i = 0; i < 4; ++i) {
      int tl = dt * 4 + i;
      int t  = tbase + tl;
      if (t < T1) {  // wave-uniform guard, EXEC stays full inside
        v8f accp, accq;
#pragma unroll
        for (int j = 0; j < 8; ++j) { accp[j] = biasP; accq[j] = biasQ; }
#pragma unroll
        for (int s = 0; s < 10; ++s) {
          int j0 = 4 * s + 2 * half;
          int ci = j0 >> 3, kk = j0 & 7;
          v2f a;
          a[0] = Xs[lane16 * XS_STRIDE + ci * WLEN + t + kk];
          a[1] = Xs[lane16 * XS_STRIDE + ci * WLEN + t + kk + 1];
          accp = wmma_f32(a, bp1[s], accp);
          accq = wmma_f32(a, bq1[s], accq);
        }
        int ch = pb * 16 + lane16;
#pragma unroll
        for (int j = 0; j < 8; ++j) {
          float hv = accp[j] * sigmoidf_(accq[j]);  // GLU in-register
          int m    = j + 8 * half;
          Hs[(tl * 16 + m) * HS_STRIDE + ch] = hv;
        }
      }
    }
    __syncthreads();
    // ---- phase 2: xt[t] = H[t] (16x64) @ Wg (64x16), one t per wave ----
    {
      int tl = wave;
      int t  = tbase + tl;
      if (t < T1) {
        v8f acc = {};
#pragma unroll
        for (int s = 0; s < 16; ++s) {
          int k0 = 4 * s + 2 * half;
          v2f a;
          a[0] = Hs[(tl * 16 + lane16) * HS_STRIDE + k0];
          a[1] = Hs[(tl * 16 + lane16) * HS_STRIDE + k0 + 1];
          acc  = wmma_f32(a, bwg[s], acc);
        }
#pragma unroll
        for (int j = 0; j < 8; ++j) {
          int m = j + 8 * half;
          xt[(t * NNODE + n0 + m) * H2C + lane16] = acc[j];
        }
      }
    }
    __syncthreads();
  }
}

// ---------------------------------------------------------------------------
// degree accumulation + rsqrt
// ---------------------------------------------------------------------------
__global__ void k_deg(const int* __restrict__ col, const float* __restrict__ ew,
                      float* __restrict__ deg) {
  int e = blockIdx.x * blockDim.x + threadIdx.x;
  if (e < NEDGE) atomicAdd(&deg[col[e]], ew[e]);
}

__global__ void k_dinv(float* __restrict__ deg) {
  int n = blockIdx.x * blockDim.x + threadIdx.x;
  if (n < NNODE) {
    float d = deg[n];
    deg[n]  = (d > 0.0f) ? rsqrtf(d) : 0.0f;
  }
}

// ---------------------------------------------------------------------------
// edge scatter: gout[t][col][f] += norm * xt[t][row][f]   (L2-resident atomics)
// one block per edge (jobs E..E+N-1 are self loops)
// ---------------------------------------------------------------------------
__global__ __launch_bounds__(256) void k_scatter(const int* __restrict__ row,
                                                 const int* __restrict__ col,
                                                 const float* __restrict__ ew,
                                                 const float* __restrict__ dinv,
                                                 const float* __restrict__ xt,
                                                 float* __restrict__ gout) {
  int job = blockIdx.x;
  int r, c;
  float w;
  if (job < NEDGE) {
    r = row[job]; c = col[job]; w = ew[job];
  } else {
    r = c = job - NEDGE; w = 1.0f;
  }
  float norm = dinv[r] * w * dinv[c];
  const float* src = xt + r * H2C;
  float* dst       = gout + c * H2C;
  for (int idx = threadIdx.x; idx < T1 * H2C; idx += 256) {
    int t = idx >> 4, f = idx & 15;
    atomicAdd(&dst[t * (NNODE * H2C) + f], norm * src[t * (NNODE * H2C) + f]);
  }
}

// ---------------------------------------------------------------------------
// conv2 (K=128 GEMM, WMMA f32) + GLU, GCN bias/relu fused into LDS load.
// one block = 16 nodes; Wc2 fragments held entirely in VGPRs (128 regs/lane)
// ---------------------------------------------------------------------------
__global__ __launch_bounds__(256) void k_conv2(const float* __restrict__ gout,
                                               const float* __restrict__ bg,
                                               const float* __restrict__ Wc2,
                                               const float* __restrict__ b2,
                                               float* __restrict__ out) {
  __shared__ float Gs[T1 * 16 * GS_STRIDE];  // 62 KB : relu(gcn+bg) [t][n][ci]

  const int tid    = threadIdx.x;
  const int lane   = tid & 31;
  const int wave   = tid >> 5;
  const int lane16 = lane & 15;
  const int half   = lane >> 4;
  const int n0     = blockIdx.x * 16;

  for (int idx = tid; idx < T1 * 256; idx += 256) {
    int t = idx >> 8, r = idx & 255;
    int n = r >> 4, ci = r & 15;
    float v = gout[(t * NNODE + n0 + n) * H2C + ci] + bg[ci];
    Gs[(t * 16 + n) * GS_STRIDE + ci] = v > 0.0f ? v : 0.0f;  // relu + bias
  }

  const int pb  = wave & 3;
  const int coP = pb * 16 + lane16;
  const int coQ = coP + 64;
  v2f bp2[32], bq2[32];  // Wc2^T fragments, K=128 in 32 steps, P+Q tiles
#pragma unroll
  for (int s = 0; s < 32; ++s) {
    int k     = 4 * s + 2 * half;
    bp2[s][0] = Wc2[coP * 128 + k];
    bp2[s][1] = Wc2[coP * 128 + k + 1];
    bq2[s][0] = Wc2[coQ * 128 + k];
    bq2[s][1] = Wc2[coQ * 128 + k + 1];
  }
  const float biasP = b2[coP];
  const float biasQ = b2[coQ];
  __syncthreads();

  const int thalf = wave >> 2;
  for (int t2 = thalf * 25; t2 < thalf * 25 + 25; ++t2) {
    v8f accp, accq;
#pragma unroll
    for (int j = 0; j < 8; ++j) { accp[j] = biasP; accq[j] = biasQ; }
#pragma unroll
    for (int s = 0; s < 32; ++s) {
      int j0 = 4 * s + 2 * half;
      int ci = j0 >> 3, kk = j0 & 7;
      v2f a;
      a[0] = Gs[((t2 + kk) * 16 + lane16) * GS_STRIDE + ci];
      a[1] = Gs[((t2 + kk + 1) * 16 + lane16) * GS_STRIDE + ci];
      accp = wmma_f32(a, bp2[s], accp);
      accq = wmma_f32(a, bq2[s], accq);
    }
#pragma unroll
    for (int j = 0; j < 8; ++j) {
      float hv = accp[j] * sigmoidf_(accq[j]);  // GLU in-register
      int m    = j + 8 * half;
      // write-once output: non-temporal, keep L2 for xt/gout working set
      __builtin_nontemporal_store(hv, &out[(n0 + m) * (H1C * T2) + coP * T2 + t2]);
    }
  }
}

// ---------------------------------------------------------------------------
extern "C" void kernel_launch(void* const* d_in, const int* in_sizes, int n_in,
                              void* d_out, int out_size, void* d_ws, size_t ws_size,
                              hipStream_t stream) {
  const float* x   = (const float*)d_in[0];
  const int*   ei  = (const int*)d_in[1];  // [2, E]
  const float* ea  = (const float*)d_in[2];
  const float* Wc1 = (const float*)d_in[4];
  const float* b1  = (const float*)d_in[5];
  const float* Wg  = (const float*)d_in[6];
  const float* bg  = (const float*)d_in[7];
  const float* Wc2 = (const float*)d_in[8];
  const float* b2  = (const float*)d_in[9];
  float*       out = (float*)d_out;

  const int* row = ei;
  const int* col = ei + NEDGE;

  const size_t xt_bytes = (size_t)T1 * NNODE * H2C * sizeof(float);  // 28.95 MB
  float* xt   = (float*)d_ws;
  float* gout = (float*)((char*)d_ws + xt_bytes);
  float* deg  = (float*)((char*)d_ws + 2 * xt_bytes);

  const int gsz = T1 * NNODE * H2C;

  k_init<<<(gsz + 255) / 256, 256, 0, stream>>>(gout, deg, gsz);
  k_conv1<<<NNODE / 16, 256, 0, stream>>>(x, Wc1, b1, Wg, xt);
  k_deg<<<(NEDGE + 255) / 256, 256, 0, stream>>>(col, ea, deg);
  k_dinv<<<(NNODE + 255) / 256, 256, 0, stream>>>(deg);
  k_scatter<<<NEDGE + NNODE, 256, 0, stream>>>(row, col, ea, deg, xt, gout);
  k_conv2<<<NNODE / 16, 256, 0, stream>>>(gout, bg, Wc2, b2, out);
}